// RNNnet_163208757728
// MI455X (gfx1250) — compile-verified
//
#include <hip/hip_runtime.h>
#include <stdint.h>

// RNN scan: h_t = tanh(x_t * w_ih + h_{t-1} @ w_hh^T), out_t = h_t . w_fo
// T = 1048576, B = 1, IN = 1, H = 3, OUT = 1.
// Strictly serial in t -> single wave32, lane 0 runs the recurrence.
// Memory is streamed through LDS with CDNA5 async-to-LDS loads (ASYNCcnt),
// double buffered, so the only cost is the per-step dependent-ALU chain.

#define CHUNK 2048                      // floats per chunk (8 KB)
#define LOADS_PER_CHUNK (CHUNK / (32 * 4))  // 16 b128 async loads per lane

__global__ __launch_bounds__(32, 1)
void rnn_scan_kernel(const float* __restrict__ x,
                     const float* __restrict__ w_ih,
                     const float* __restrict__ w_hh,
                     const float* __restrict__ w_fo,
                     float* __restrict__ out,
                     int T)
{
    __shared__ __attribute__((aligned(16))) float xbuf[2 * CHUNK];
    __shared__ __attribute__((aligned(16))) float obuf[CHUNK];

    const int lane = threadIdx.x;          // wave32: 0..31
    const int nch  = T / CHUNK;

    // Uniform weights (compiler emits scalar loads).
    const float wi0 = w_ih[0], wi1 = w_ih[1], wi2 = w_ih[2];
    const float a00 = w_hh[0], a01 = w_hh[1], a02 = w_hh[2];
    const float a10 = w_hh[3], a11 = w_hh[4], a12 = w_hh[5];
    const float a20 = w_hh[6], a21 = w_hh[7], a22 = w_hh[8];
    const float f0  = w_fo[0], f1  = w_fo[1], f2  = w_fo[2];

    // Issue one chunk of async global->LDS b128 loads (16 per lane).
    auto issue_chunk = [&](int c) {
        const float* gp = x + (size_t)c * CHUNK;
        uint32_t base = (uint32_t)(uintptr_t)(&xbuf[(c & 1) * CHUNK]);
#pragma unroll
        for (int k = 0; k < LOADS_PER_CHUNK; ++k) {
            uint32_t    loff = base + (uint32_t)((k * 32 + lane) * 16);
            const float* g  = gp + (k * 32 + lane) * 4;
            asm volatile("global_load_async_to_lds_b128 %0, %1, off"
                         :: "v"(loff), "v"(g)
                         : "memory");
        }
    };

    float h0 = 0.0f, h1 = 0.0f, h2 = 0.0f;

    if (nch > 0) {
        issue_chunk(0);

        for (int c = 0; c < nch; ++c) {
            if (c + 1 < nch) {
                issue_chunk(c + 1);  // prefetch next chunk into other buffer
                // In-order async completion: <=16 outstanding => chunk c done.
                asm volatile("s_wait_asynccnt 16" ::: "memory");
            } else {
                asm volatile("s_wait_asynccnt 0" ::: "memory");
            }

            if (lane == 0) {
                const float* xb = &xbuf[(c & 1) * CHUNK];
                for (int t = 0; t < CHUNK; t += 4) {
                    float4 xv = *(const float4*)(xb + t);   // ds_load_b128
                    float xs[4] = {xv.x, xv.y, xv.z, xv.w};
                    float ov[4];
#pragma unroll
                    for (int j = 0; j < 4; ++j) {
                        float xx = xs[j];
                        // Critical path: 3 chained FMA + 1 hardware tanh.
                        float t0 = fmaf(h2, a02, fmaf(h1, a01, fmaf(h0, a00, xx * wi0)));
                        float t1 = fmaf(h2, a12, fmaf(h1, a11, fmaf(h0, a10, xx * wi1)));
                        float t2 = fmaf(h2, a22, fmaf(h1, a21, fmaf(h0, a20, xx * wi2)));
#if __has_builtin(__builtin_amdgcn_tanhf)
                        h0 = __builtin_amdgcn_tanhf(t0);
                        h1 = __builtin_amdgcn_tanhf(t1);
                        h2 = __builtin_amdgcn_tanhf(t2);
#else
                        // gfx1250 TRANS op; v_nop covers the ISA's
                        // "1 independent op after TRANS" hazard for the last one.
                        asm("v_tanh_f32 %0, %3\n\t"
                            "v_tanh_f32 %1, %4\n\t"
                            "v_tanh_f32 %2, %5\n\t"
                            "v_nop"
                            : "=&v"(h0), "=&v"(h1), "=&v"(h2)
                            : "v"(t0), "v"(t1), "v"(t2));
#endif
                        ov[j] = fmaf(h2, f2, fmaf(h1, f1, h0 * f0));
                    }
                    *(float4*)(&obuf[t]) = make_float4(ov[0], ov[1], ov[2], ov[3]);
                }
            }
            __syncthreads();

            // Coalesced cooperative store: 512 B per wave instruction.
            float* og = out + (size_t)c * CHUNK;
#pragma unroll
            for (int k = 0; k < LOADS_PER_CHUNK; ++k) {
                int idx = (k * 32 + lane) * 4;
                float4 v = *(const float4*)(&obuf[idx]);
                *(float4*)(og + idx) = v;
            }
            __syncthreads();
        }
    }

    // Tail (T not a multiple of CHUNK; not hit for T = 2^20, kept for safety).
    if (lane == 0) {
        for (int t = nch * CHUNK; t < T; ++t) {
            float xx = x[t];
            float t0 = fmaf(h2, a02, fmaf(h1, a01, fmaf(h0, a00, xx * wi0)));
            float t1 = fmaf(h2, a12, fmaf(h1, a11, fmaf(h0, a10, xx * wi1)));
            float t2 = fmaf(h2, a22, fmaf(h1, a21, fmaf(h0, a20, xx * wi2)));
            h0 = tanhf(t0); h1 = tanhf(t1); h2 = tanhf(t2);
            out[t] = fmaf(h2, f2, fmaf(h1, f1, h0 * f0));
        }
    }
}

extern "C" void kernel_launch(void* const* d_in, const int* in_sizes, int n_in,
                              void* d_out, int out_size, void* d_ws, size_t ws_size,
                              hipStream_t stream) {
    const float* x    = (const float*)d_in[0];   // [T,1,1]
    const float* w_ih = (const float*)d_in[1];   // [3,1]
    const float* w_hh = (const float*)d_in[2];   // [3,3]
    const float* w_fo = (const float*)d_in[3];   // [1,3]
    float*       out  = (float*)d_out;           // [T,1,1]
    const int T = in_sizes[0];

    hipLaunchKernelGGL(rnn_scan_kernel, dim3(1), dim3(32), 0, stream,
                       x, w_ih, w_hh, w_fo, out, T);
}